// GraphNeuralNetwork_56788057587942
// MI455X (gfx1250) — compile-verified
//
#include <hip/hip_runtime.h>
#include <math.h>

#define BB 8
#define NN 512
#define FF 32
#define HH 64
#define LL 2

typedef float v2f __attribute__((ext_vector_type(2)));
typedef float v8f __attribute__((ext_vector_type(8)));

// ---------------------------------------------------------------------------
// K1: h = relu(x@W_enc + b_enc); zi = h@Wa; zj = h@Wb
// one 64-thread block per (b,n) row
// ---------------------------------------------------------------------------
__global__ void gnn_encode(const float* __restrict__ x,
                           const float* __restrict__ W_enc,
                           const float* __restrict__ b_enc,
                           const float* __restrict__ W_e1,
                           float* __restrict__ h,
                           float* __restrict__ zi,
                           float* __restrict__ zj) {
    int row = blockIdx.x;            // 0 .. B*N-1
    int t   = threadIdx.x;           // 0 .. 63
    __shared__ float xs[FF];
    __shared__ float hs[HH];
    if (t < FF) xs[t] = x[row * FF + t];
    __syncthreads();
    float acc = b_enc[t];
#pragma unroll
    for (int f = 0; f < FF; ++f) acc = fmaf(xs[f], W_enc[f * HH + t], acc);
    float hv = fmaxf(acc, 0.0f);
    hs[t] = hv;
    h[row * HH + t] = hv;
    __syncthreads();
    float za = 0.0f, zb = 0.0f;
#pragma unroll
    for (int k = 0; k < HH; ++k) {
        float hk = hs[k];
        za = fmaf(hk, W_e1[k * HH + t], za);           // Wa = W_e1[:H]
        zb = fmaf(hk, W_e1[(HH + k) * HH + t], zb);    // Wb = W_e1[H:]
    }
    zi[row * HH + t] = za;
    zj[row * HH + t] = zb;
}

// ---------------------------------------------------------------------------
// K2: logits[b,i,j] = sum_h relu(zi+zj+b_e1)*W_e2 + b_e2, diag=-inf, softmax
// one 256-thread block per (b,i); each thread handles j and j+256
// ---------------------------------------------------------------------------
__global__ void gnn_adj(const float* __restrict__ zi,
                        const float* __restrict__ zj,
                        const float* __restrict__ b_e1,
                        const float* __restrict__ W_e2,
                        const float* __restrict__ b_e2,
                        float* __restrict__ adj) {
    int b = blockIdx.x / NN;
    int i = blockIdx.x % NN;
    int t = threadIdx.x;             // 0 .. 255
    __shared__ float zis[HH], be1s[HH], we2s[HH];
    __shared__ float red[256];
    if (t < HH) {
        zis[t]  = zi[((size_t)b * NN + i) * HH + t];
        be1s[t] = b_e1[t];
        we2s[t] = W_e2[t];
    }
    __syncthreads();
    float l0[2];
#pragma unroll
    for (int s = 0; s < 2; ++s) {
        int j = t + s * 256;
        const float* zr = &zj[((size_t)b * NN + j) * HH];
        float a = 0.0f;
#pragma unroll 8
        for (int k = 0; k < HH; ++k) {
            float v = zis[k] + zr[k] + be1s[k];
            a = fmaf(fmaxf(v, 0.0f), we2s[k], a);
        }
        a += b_e2[0];
        if (j == i) a = -INFINITY;
        l0[s] = a;
    }
    // block max
    red[t] = fmaxf(l0[0], l0[1]);
    __syncthreads();
    for (int s2 = 128; s2 > 0; s2 >>= 1) {
        if (t < s2) red[t] = fmaxf(red[t], red[t + s2]);
        __syncthreads();
    }
    float mx = red[0];
    __syncthreads();
    float e0 = expf(l0[0] - mx);
    float e1 = expf(l0[1] - mx);
    red[t] = e0 + e1;
    __syncthreads();
    for (int s2 = 128; s2 > 0; s2 >>= 1) {
        if (t < s2) red[t] += red[t + s2];
        __syncthreads();
    }
    float inv = 1.0f / red[0];
    float* arow = adj + ((size_t)b * NN + i) * NN;
    arow[t]       = e0 * inv;
    arow[t + 256] = e1 * inv;
}

// ---------------------------------------------------------------------------
// K3: m = adj @ h  (batched 512x512 @ 512x64) with V_WMMA_F32_16X16X4_F32.
// grid = B*(N/16) blocks of 128 threads; wave w owns n-tile w (N dim = 64).
// Fragment layouts per CDNA5 ISA 7.12.2 (32-bit A 16x4, f32 C/D 16x16).
// ---------------------------------------------------------------------------
__global__ void gnn_msg_wmma(const float* __restrict__ adj,
                             const float* __restrict__ h,
                             float* __restrict__ m) {
    int blk  = blockIdx.x;
    int b    = blk >> 5;                 // / (N/16)
    int i0   = (blk & 31) << 4;
    int wave = threadIdx.x >> 5;
    int lane = threadIdx.x & 31;
    int n0   = wave << 4;

    const float* A  = adj + (size_t)b * NN * NN;
    const float* Hm = h   + (size_t)b * NN * HH;

    int arow = i0 + (lane & 15);
    int kk   = (lane & 16) >> 3;         // 0 for lanes 0-15, 2 for lanes 16-31
    int ncol = n0 + (lane & 15);

    v8f acc = {};
    for (int k = 0; k < NN; k += 4) {
        v2f a = *(const v2f*)(A + (size_t)arow * NN + k + kk);
        v2f bf;
        bf.x = Hm[(k + kk) * HH + ncol];
        bf.y = Hm[(k + kk + 1) * HH + ncol];
        acc = __builtin_amdgcn_wmma_f32_16x16x4_f32(
            /*neg_a=*/false, a, /*neg_b=*/false, bf,
            /*c_mod=*/(short)0, acc, /*reuse_a=*/false, /*reuse_b=*/false);
    }

    float* M = m + (size_t)b * NN * HH;
    int rbase = i0 + ((lane & 16) >> 1); // +8 for upper half-wave
#pragma unroll
    for (int r = 0; r < 8; ++r) {
        M[(rbase + r) * HH + ncol] = acc[r];
    }
}

// ---------------------------------------------------------------------------
// K4: fused msg-MLP + GRU cell. one 192-thread block per (b,n) row.
// ---------------------------------------------------------------------------
__global__ void gnn_gru(const float* __restrict__ m,
                        const float* __restrict__ h,
                        const float* __restrict__ W_msg,
                        const float* __restrict__ b_msg,
                        const float* __restrict__ W_ih,
                        const float* __restrict__ b_ih,
                        const float* __restrict__ W_hh,
                        const float* __restrict__ b_hh,
                        float* __restrict__ h_out) {
    int row = blockIdx.x;
    int t   = threadIdx.x;           // 0 .. 191
    __shared__ float ms[HH], hs[HH], m2s[HH], gis[3 * HH], ghs[3 * HH];
    if (t < HH) {
        ms[t] = m[(size_t)row * HH + t];
        hs[t] = h[(size_t)row * HH + t];
    }
    __syncthreads();
    if (t < HH) {
        float a = b_msg[t];
#pragma unroll 8
        for (int k = 0; k < HH; ++k) a = fmaf(ms[k], W_msg[k * HH + t], a);
        m2s[t] = fmaxf(a, 0.0f);
    }
    __syncthreads();
    {
        float gi = b_ih[t], gh = b_hh[t];
#pragma unroll 8
        for (int k = 0; k < HH; ++k) {
            gi = fmaf(m2s[k], W_ih[k * (3 * HH) + t], gi);
            gh = fmaf(hs[k],  W_hh[k * (3 * HH) + t], gh);
        }
        gis[t] = gi;
        ghs[t] = gh;
    }
    __syncthreads();
    if (t < HH) {
        float r = 1.0f / (1.0f + expf(-(gis[t] + ghs[t])));
        float z = 1.0f / (1.0f + expf(-(gis[t + HH] + ghs[t + HH])));
        float n = tanhf(gis[t + 2 * HH] + r * ghs[t + 2 * HH]);
        h_out[(size_t)row * HH + t] = (1.0f - z) * n + z * hs[t];
    }
}

// ---------------------------------------------------------------------------
// K5: out[b] = mean_n(h[b,n,:]) @ W_out + b_out
// ---------------------------------------------------------------------------
__global__ void gnn_readout(const float* __restrict__ h,
                            const float* __restrict__ W_out,
                            const float* __restrict__ b_out,
                            float* __restrict__ out) {
    int b = blockIdx.x;
    int t = threadIdx.x;             // 0 .. 63
    __shared__ float red[HH];
    float s = 0.0f;
    for (int n = 0; n < NN; ++n) s += h[((size_t)b * NN + n) * HH + t];
    red[t] = (s * (1.0f / NN)) * W_out[t];
    __syncthreads();
    for (int st = 32; st > 0; st >>= 1) {
        if (t < st) red[t] += red[t + st];
        __syncthreads();
    }
    if (t == 0) out[b] = red[0] + b_out[0];
}

// ---------------------------------------------------------------------------
extern "C" void kernel_launch(void* const* d_in, const int* in_sizes, int n_in,
                              void* d_out, int out_size, void* d_ws, size_t ws_size,
                              hipStream_t stream) {
    (void)in_sizes; (void)n_in; (void)out_size; (void)ws_size;
    const float* x     = (const float*)d_in[0];
    const float* W_enc = (const float*)d_in[1];
    const float* b_enc = (const float*)d_in[2];
    const float* W_e1  = (const float*)d_in[3];
    const float* b_e1  = (const float*)d_in[4];
    const float* W_e2  = (const float*)d_in[5];
    const float* b_e2  = (const float*)d_in[6];
    const float* W_msg = (const float*)d_in[7];
    const float* b_msg = (const float*)d_in[8];
    const float* W_ih  = (const float*)d_in[9];
    const float* b_ih  = (const float*)d_in[10];
    const float* W_hh  = (const float*)d_in[11];
    const float* b_hh  = (const float*)d_in[12];
    const float* W_out = (const float*)d_in[13];
    const float* b_out = (const float*)d_in[14];
    float* out = (float*)d_out;

    float* ws = (float*)d_ws;
    const size_t BNH = (size_t)BB * NN * HH;
    float* h0   = ws;                 // B*N*H
    float* h1   = h0 + BNH;           // B*N*H
    float* zi   = h1 + BNH;           // B*N*H
    float* zj   = zi + BNH;           // B*N*H
    float* adj  = zj + BNH;           // B*N*N
    float* mbuf = adj + (size_t)BB * NN * NN;  // B*N*H   (total ~13.6 MB)

    gnn_encode<<<BB * NN, HH, 0, stream>>>(x, W_enc, b_enc, W_e1, h0, zi, zj);
    gnn_adj<<<BB * NN, 256, 0, stream>>>(zi, zj, b_e1, W_e2, b_e2, adj);

    float* hc = h0;
    float* hn = h1;
    for (int l = 0; l < LL; ++l) {
        gnn_msg_wmma<<<BB * (NN / 16), 128, 0, stream>>>(adj, hc, mbuf);
        gnn_gru<<<BB * NN, 3 * HH, 0, stream>>>(
            mbuf, hc,
            W_msg + (size_t)l * HH * HH,     b_msg + (size_t)l * HH,
            W_ih  + (size_t)l * HH * 3 * HH, b_ih  + (size_t)l * 3 * HH,
            W_hh  + (size_t)l * HH * 3 * HH, b_hh  + (size_t)l * 3 * HH,
            hn);
        float* tmp = hc; hc = hn; hn = tmp;
    }
    gnn_readout<<<BB, HH, 0, stream>>>(hc, W_out, b_out, out);
}